// SE2CGNonLinearity_6820408066528
// MI455X (gfx1250) — compile-verified
//
#include <hip/hip_runtime.h>

typedef __attribute__((ext_vector_type(2))) float v2f;
typedef __attribute__((ext_vector_type(8))) float v8f;

#define PLANE   131072      // O*X*Y = 32*64*64
#define BSTRIDE 1048576     // L*PLANE
#define NPAIR   20
#define NCHUNK  5           // K = 20 -> 5 chunks of 4
#define TILES_PER_WAVE 8    // 65536 tiles / 8192 waves

// out[b,l,p] = sum_{m<=n, m+n=l} coef(l,m,n) * x_m * x_n   (complex)
// coef = w[l,m] (+ w[l,n] if m != n).
// D1 = [A_re; A_im] @ P_re   -> lanes 0-15: A_re@P_re, lanes 16-31: A_im@P_re
// D2 = [A_im; A_re] @ P_im   -> lanes 0-15: A_im@P_im, lanes 16-31: A_re@P_im
// out_re = D1 - D2 (lanes 0-15), out_im = D1 + D2 (lanes 16-31): no shuffles.

__global__ __launch_bounds__(256) void se2cg_wmma_kernel(
    const float* __restrict__ xre, const float* __restrict__ xim,
    const float* __restrict__ wre, const float* __restrict__ wim,
    float* __restrict__ out)
{
    // unordered pairs {m,n}, m<=n, m+n<8  (20 of them)
    constexpr int PM[NPAIR] = {0,0,0,0,0,0,0,0, 1,1,1,1,1,1, 2,2,2,2, 3,3};
    constexpr int PN[NPAIR] = {0,1,2,3,4,5,6,7, 1,2,3,4,5,6, 2,3,4,5, 3,4};

    const int lane      = threadIdx.x & 31;
    const int waveInBlk = threadIdx.x >> 5;
    const int wave      = blockIdx.x * 8 + waveInBlk;
    const int col       = lane & 15;        // pixel column within 16-wide tile
    const bool hi       = lane >= 16;       // upper lane half

    // ---- A fragments (constant across pixels; computed once per wave) ----
    // A-matrix layout: lane owns row M = lane&15; VGPR0/1 hold K = {0,1} for
    // lanes 0-15 and K = {2,3} for lanes 16-31 within each K-chunk of 4.
    const int   M     = lane & 15;
    const int   l     = M & 7;
    const float* wA = (M < 8) ? wre : wim;   // stack [A_re; A_im]
    const float* wB = (M < 8) ? wim : wre;   // stack [A_im; A_re]
    float afA0[NCHUNK], afA1[NCHUNK], afB0[NCHUNK], afB1[NCHUNK];
#pragma unroll
    for (int kk = 0; kk < NCHUNK; ++kk) {
        float vA[4], vB[4];
#pragma unroll
        for (int q = 0; q < 4; ++q) {
            const int j = 4 * kk + q;
            const int m = PM[j], n = PN[j];
            float cA = wA[l * 8 + m];
            float cB = wB[l * 8 + m];
            if (m != n) { cA += wA[l * 8 + n]; cB += wB[l * 8 + n]; }
            const bool on = (l == m + n);
            vA[q] = on ? cA : 0.0f;
            vB[q] = on ? cB : 0.0f;
        }
        afA0[kk] = hi ? vA[2] : vA[0];
        afA1[kk] = hi ? vA[3] : vA[1];
        afB0[kk] = hi ? vB[2] : vB[0];
        afB1[kk] = hi ? vB[3] : vB[1];
    }

    for (int t = 0; t < TILES_PER_WAVE; ++t) {
        const int tile = wave * TILES_PER_WAVE + t;   // 0..65535, full coverage
        const int b    = tile >> 13;                  // tile / 8192
        const int p0   = (tile & 8191) << 4;          // 16-pixel tile origin
        const int p    = p0 + col;

        // ---- load x for this lane's pixel (duplicated across lane halves) ----
        float xr[8], xi[8];
        const long xb = (long)b * BSTRIDE + p;
#pragma unroll
        for (int m = 0; m < 8; ++m) {
            xr[m] = xre[xb + (long)m * PLANE];
            xi[m] = xim[xb + (long)m * PLANE];
        }

        // ---- K-loop: 5 chunks x {P_re, P_im} WMMAs ----
        v8f d1 = {};   // [A_re; A_im] @ P_re
        v8f d2 = {};   // [A_im; A_re] @ P_im
#pragma unroll
        for (int kk = 0; kk < NCHUNK; ++kk) {
            float pr[4], pi[4];
#pragma unroll
            for (int q = 0; q < 4; ++q) {
                const int j = 4 * kk + q;
                const int m = PM[j], n = PN[j];
                pr[q] = xr[m] * xr[n] - xi[m] * xi[n];
                pi[q] = xr[m] * xi[n] + xi[m] * xr[n];
            }
            // B-matrix layout: VGPR0 = K row 0 (lanes 0-15) / row 2 (16-31),
            //                  VGPR1 = K row 1 / row 3.
            v2f aA;  aA.x  = afA0[kk];           aA.y  = afA1[kk];
            v2f aB;  aB.x  = afB0[kk];           aB.y  = afB1[kk];
            v2f bre; bre.x = hi ? pr[2] : pr[0]; bre.y = hi ? pr[3] : pr[1];
            v2f bim; bim.x = hi ? pi[2] : pi[0]; bim.y = hi ? pi[3] : pi[1];
            d1 = __builtin_amdgcn_wmma_f32_16x16x4_f32(
                     false, aA, false, bre, (short)0, d1, false, false);
            d2 = __builtin_amdgcn_wmma_f32_16x16x4_f32(
                     false, aB, false, bim, (short)0, d2, false, false);
        }

        // ---- combine (same-lane) and store (interleaved complex64) ----
        const long obase = 2l * ((long)(b * 8) * PLANE + p) + (hi ? 1 : 0);
#pragma unroll
        for (int r = 0; r < 8; ++r) {
            float val = hi ? (d1[r] + d2[r]) : (d1[r] - d2[r]);
            out[obase + 2l * (long)r * PLANE] = val; // wave: 128B contiguous
        }
    }
}

extern "C" void kernel_launch(void* const* d_in, const int* in_sizes, int n_in,
                              void* d_out, int out_size, void* d_ws, size_t ws_size,
                              hipStream_t stream) {
    const float* xre = (const float*)d_in[0];
    const float* xim = (const float*)d_in[1];
    const float* wre = (const float*)d_in[2];
    const float* wim = (const float*)d_in[3];
    // d_in[4] (cg) is baked into the pair table / selection rule.
    float* out = (float*)d_out;
    // 1024 blocks x 256 threads = 8192 waves; 8 tiles/wave = 65536 tiles
    // = B(8) * 8192 sixteen-pixel tiles -> exact coverage, no bounds checks.
    se2cg_wmma_kernel<<<1024, 256, 0, stream>>>(xre, xim, wre, wim, out);
}